// Bending_42880953484261
// MI455X (gfx1250) — compile-verified
//
#include <hip/hip_runtime.h>
#include <math.h>

// ARAP local step for MI455X (gfx1250, wave32).
// Covariance segment-sum via V_WMMA_F32_16X16X4_F32 (full f32 precision):
//   tile of 5 vertices per wave; row m=3v+r <- w*d1[r], col n=3v+c <- d2[c],
//   K = neighbor slot (16 neighbors -> 4 chained K=4 WMMAs). Diagonal 3x3
//   blocks of the 16x16 accumulator are each vertex's covariance S.
// LDS staging is row-major [row=3v+r][j] with a zeroed row 15 so fragment
// loads are unconditional ds_load_b64 (no per-lane exec predication).
// Rotation: Mueller et al. quaternion iteration on A = S^T, which converges to
//   argmax_{R in SO(3)} tr(R^T S^T) == reference V*diag(1,1,detfix)*U^T.

typedef float v2f __attribute__((ext_vector_type(2)));
typedef float v8f __attribute__((ext_vector_type(8)));

#define WAVES_PER_BLOCK 8
#define VPT   5    // vertices per wave tile (3*5=15 rows/cols of the 16x16 tile)
#define MAXNB 16   // neighbor slots handled per vertex (reference uses D=16)

__device__ __forceinline__ void quat_to_mat(float q0, float q1, float q2, float q3,
                                            float R[3][3]) {
    float xx = q1*q1, yy = q2*q2, zz = q3*q3;
    float xy = q1*q2, xz = q1*q3, yz = q2*q3;
    float wx = q0*q1, wy = q0*q2, wz = q0*q3;
    R[0][0] = 1.0f - 2.0f*(yy+zz); R[0][1] = 2.0f*(xy-wz);       R[0][2] = 2.0f*(xz+wy);
    R[1][0] = 2.0f*(xy+wz);        R[1][1] = 1.0f - 2.0f*(xx+zz); R[1][2] = 2.0f*(yz-wx);
    R[2][0] = 2.0f*(xz-wy);        R[2][1] = 2.0f*(yz+wx);       R[2][2] = 1.0f - 2.0f*(xx+yy);
}

__global__ __launch_bounds__(256) void arap_rotations_kernel(
    const float* __restrict__ xyz1, const float* __restrict__ xyz2,
    const int*   __restrict__ nbrList, const int* __restrict__ numNbr,
    const int*   __restrict__ accNbr,  const float* __restrict__ wgt,
    float* __restrict__ out,
    int B, int N, int tilesPerBatch, int totalWaves)
{
    // Row-major 16x16 staging tiles: row = 3v+r (row 15 = zero pad), col = j.
    __shared__ float sA[WAVES_PER_BLOCK][16 * 16]; // w * (x1_i - x1_j)
    __shared__ float sB[WAVES_PER_BLOCK][16 * 16]; // (x2_i - x2_j)
    __shared__ float sC[WAVES_PER_BLOCK][16 * 16]; // 16x16 accumulator tile

    const int lane = threadIdx.x & 31;
    const int wv   = threadIdx.x >> 5;
    const int waveGlobal = blockIdx.x * WAVES_PER_BLOCK + wv;
    const bool waveValid = waveGlobal < totalWaves;
    const int wg   = waveValid ? waveGlobal : 0;
    const int b    = wg / tilesPerBatch;
    const int tile = wg % tilesPerBatch;
    const int vbase = tile * VPT;

    // ---- Phase 1: gather edges, write w*d1 and d2 into LDS -----------------
    // Zero the pad row (row 15) so lane 15's fragment reads are valid zeros.
    if (lane < 16) {
        sA[wv][15 * 16 + lane] = 0.0f;
        sB[wv][15 * 16 + lane] = 0.0f;
    }
    for (int t = 0; t < 3; ++t) {
        int e = t * 32 + lane;                 // 0..79 valid (5 verts * 16 nbrs)
        if (e < VPT * MAXNB) {
            int v = e / MAXNB, j = e - v * MAXNB;
            int i = vbase + v;
            float a0 = 0.f, a1 = 0.f, a2 = 0.f, b0 = 0.f, b1 = 0.f, b2 = 0.f;
            if (waveValid && i < N && j < numNbr[i]) {
                int eo = accNbr[i] + j;
                int jn = nbrList[eo];
                float w = wgt[eo];
                const float* p1i = xyz1 + ((size_t)b * N + i)  * 3;
                const float* p1j = xyz1 + ((size_t)b * N + jn) * 3;
                const float* p2i = xyz2 + ((size_t)b * N + i)  * 3;
                const float* p2j = xyz2 + ((size_t)b * N + jn) * 3;
                a0 = w * (p1i[0] - p1j[0]);
                a1 = w * (p1i[1] - p1j[1]);
                a2 = w * (p1i[2] - p1j[2]);
                b0 = p2i[0] - p2j[0];
                b1 = p2i[1] - p2j[1];
                b2 = p2i[2] - p2j[2];
            }
            int row0 = 3 * v;
            sA[wv][(row0 + 0) * 16 + j] = a0;
            sA[wv][(row0 + 1) * 16 + j] = a1;
            sA[wv][(row0 + 2) * 16 + j] = a2;
            sB[wv][(row0 + 0) * 16 + j] = b0;
            sB[wv][(row0 + 1) * 16 + j] = b1;
            sB[wv][(row0 + 2) * 16 + j] = b2;
        }
    }
    __syncthreads();

    // ---- Phase 2: covariance via V_WMMA_F32_16X16X4_F32 --------------------
    // A 16x4 layout (ISA 7.12.2): V0: K=0 (lanes 0-15) / K=2 (lanes 16-31);
    //                             V1: K=1 / K=3.  M = lane & 15.
    // B 4x16 assumed symmetric:   V0: K=0 / K=2;  V1: K=1 / K=3.  N = lane & 15.
    const int half = lane >> 4;
    const int lm   = lane & 15;
    // Row index is just lm (row 15 is the zero pad row) -> unconditional loads.
    const v2f* rowA = (const v2f*)&sA[wv][lm * 16];
    const v2f* rowB = (const v2f*)&sB[wv][lm * 16];

    v8f c = {};
    #pragma unroll
    for (int chunk = 0; chunk < 4; ++chunk) {
        int p = chunk * 2 + half;              // pair index: j = {2p, 2p+1}
        v2f a  = rowA[p];                      // ds_load_b64
        v2f bb = rowB[p];                      // ds_load_b64
        c = __builtin_amdgcn_wmma_f32_16x16x4_f32(
                /*neg_a=*/false, a, /*neg_b=*/false, bb,
                /*c_mod=*/(short)0, c, /*reuse_a=*/false, /*reuse_b=*/false);
    }
    // Spill 16x16 C tile: lane half h holds rows 8h..8h+7, col = lm.
    #pragma unroll
    for (int r = 0; r < 8; ++r)
        sC[wv][(half * 8 + r) * 16 + lm] = c[r];
    __syncthreads();

    // ---- Phase 3: rotation extraction (one lane per vertex) ----------------
    if (waveValid && lane < VPT) {
        int i = vbase + lane;
        if (i < N) {
            // At = S^T ; S[r][c] = C[3v+r][3v+c]
            float At[3][3];
            #pragma unroll
            for (int r = 0; r < 3; ++r)
                #pragma unroll
                for (int cc = 0; cc < 3; ++cc)
                    At[r][cc] = sC[wv][(3*lane + cc) * 16 + (3*lane + r)];

            float q0 = 1.f, q1 = 0.f, q2 = 0.f, q3 = 0.f;
            #pragma unroll 1
            for (int it = 0; it < 30; ++it) {
                float R[3][3];
                quat_to_mat(q0, q1, q2, q3, R);
                float ox = 0.f, oy = 0.f, oz = 0.f, d = 0.f;
                #pragma unroll
                for (int j = 0; j < 3; ++j) {
                    float rx = R[0][j], ry = R[1][j], rz = R[2][j];
                    float ax = At[0][j], ay = At[1][j], az = At[2][j];
                    ox += ry * az - rz * ay;
                    oy += rz * ax - rx * az;
                    oz += rx * ay - ry * ax;
                    d  += rx * ax + ry * ay + rz * az;
                }
                float inv = 1.0f / (fabsf(d) + 1.0e-9f);
                ox *= inv; oy *= inv; oz *= inv;
                float ang  = sqrtf(ox*ox + oy*oy + oz*oz);
                float hl   = 0.5f * ang;
                float k    = sinf(hl) / fmaxf(ang, 1.0e-12f); // -> 0.5 as ang->0
                float dqw  = cosf(hl);
                float dqx = ox * k, dqy = oy * k, dqz = oz * k;
                float nw = dqw*q0 - dqx*q1 - dqy*q2 - dqz*q3;
                float nx = dqw*q1 + dqx*q0 + dqy*q3 - dqz*q2;
                float ny = dqw*q2 - dqx*q3 + dqy*q0 + dqz*q1;
                float nz = dqw*q3 + dqx*q2 - dqy*q1 + dqz*q0;
                float invn = rsqrtf(nw*nw + nx*nx + ny*ny + nz*nz);
                q0 = nw * invn; q1 = nx * invn; q2 = ny * invn; q3 = nz * invn;
            }
            float R[3][3];
            quat_to_mat(q0, q1, q2, q3, R);
            float* po = out + ((size_t)b * N + i) * 9;
            #pragma unroll
            for (int r = 0; r < 3; ++r)
                #pragma unroll
                for (int cc = 0; cc < 3; ++cc)
                    po[r * 3 + cc] = R[r][cc];
        }
    }
}

extern "C" void kernel_launch(void* const* d_in, const int* in_sizes, int n_in,
                              void* d_out, int out_size, void* d_ws, size_t ws_size,
                              hipStream_t stream) {
    (void)n_in; (void)d_ws; (void)ws_size; (void)out_size;
    const float* xyz1 = (const float*)d_in[0];
    const float* xyz2 = (const float*)d_in[1];
    const int*   nbr  = (const int*)  d_in[2];
    const int*   nnum = (const int*)  d_in[3];
    const int*   acc  = (const int*)  d_in[4];
    const float* wgt  = (const float*)d_in[5];
    // d_in[6] = rotations (identity init, matches our identity quaternion seed)
    // d_in[7] = arapWeight (does not affect fitted rotations)

    const int N = in_sizes[3];                  // numNeighbors has shape (N,)
    const int B = in_sizes[0] / (3 * N);        // xyz1 is B*N*3
    const int tilesPerBatch = (N + VPT - 1) / VPT;
    const int totalWaves = B * tilesPerBatch;
    const int blocks = (totalWaves + WAVES_PER_BLOCK - 1) / WAVES_PER_BLOCK;

    arap_rotations_kernel<<<blocks, WAVES_PER_BLOCK * 32, 0, stream>>>(
        xyz1, xyz2, nbr, nnum, acc, wgt, (float*)d_out,
        B, N, tilesPerBatch, totalWaves);
}